// DistinctionLoss_72284299592193
// MI455X (gfx1250) — compile-verified
//
#include <hip/hip_runtime.h>
#include <hip/hip_bf16.h>
#include <math.h>

// Problem constants (fixed by the reference setup_inputs()).
#define BB 8
#define NN 4096
#define DD 256
#define EPS_NORM 1e-12f
#define LOG_CLAMP -100.0f

typedef float v2f __attribute__((ext_vector_type(2)));
typedef float v8f __attribute__((ext_vector_type(8)));

// Workspace layout (floats):
//   sPart  : [B][32][256]  = 65536     per-block partial sums of normalized rows
//   sBuf   : [B][256]      = 2048      s_b = sum over N of normalized rows
//   bcePart: [2048]                    per-wave BCE partial sums
//   invnorm: [B*N]         = 32768     1 / max(||row||, eps)
#define WS_SPART   0
#define WS_SBUF    65536
#define WS_BCEPART 67584
#define WS_INVNORM 69632

// ---------------------------------------------------------------------------
// Kernel 1: per-row inverse norms + per-block partial column sums of f.
// grid (32, B), block 256 (8 waves). Each wave handles 16 rows.
// ---------------------------------------------------------------------------
__global__ void dl_pass1(const float* __restrict__ features,
                         float* __restrict__ sPart,
                         float* __restrict__ invnorm) {
  __shared__ float red[8 * 256];
  const int lane = threadIdx.x & 31;
  const int w    = threadIdx.x >> 5;
  const int b    = blockIdx.y;
  const int r0   = (blockIdx.x * 8 + w) * 16;
  const float* fB = features + (size_t)b * NN * DD;

  float ps0x = 0.f, ps0y = 0.f, ps0z = 0.f, ps0w = 0.f;
  float ps1x = 0.f, ps1y = 0.f, ps1z = 0.f, ps1w = 0.f;

  for (int rr = 0; rr < 16; ++rr) {
    const int row = r0 + rr;
    const float4 x0 = *(const float4*)(fB + (size_t)row * DD + lane * 4);
    const float4 x1 = *(const float4*)(fB + (size_t)row * DD + 128 + lane * 4);
    float ssq = x0.x * x0.x + x0.y * x0.y + x0.z * x0.z + x0.w * x0.w +
                x1.x * x1.x + x1.y * x1.y + x1.z * x1.z + x1.w * x1.w;
    // full wave32 reduction
    ssq += __shfl_xor(ssq, 16, 32);
    ssq += __shfl_xor(ssq, 8, 32);
    ssq += __shfl_xor(ssq, 4, 32);
    ssq += __shfl_xor(ssq, 2, 32);
    ssq += __shfl_xor(ssq, 1, 32);
    const float inv = 1.0f / fmaxf(sqrtf(ssq), EPS_NORM);
    if (lane == 0) invnorm[b * NN + row] = inv;
    ps0x += x0.x * inv; ps0y += x0.y * inv; ps0z += x0.z * inv; ps0w += x0.w * inv;
    ps1x += x1.x * inv; ps1y += x1.y * inv; ps1z += x1.z * inv; ps1w += x1.w * inv;
  }
  red[w * 256 + lane * 4 + 0] = ps0x;
  red[w * 256 + lane * 4 + 1] = ps0y;
  red[w * 256 + lane * 4 + 2] = ps0z;
  red[w * 256 + lane * 4 + 3] = ps0w;
  red[w * 256 + 128 + lane * 4 + 0] = ps1x;
  red[w * 256 + 128 + lane * 4 + 1] = ps1y;
  red[w * 256 + 128 + lane * 4 + 2] = ps1z;
  red[w * 256 + 128 + lane * 4 + 3] = ps1w;
  __syncthreads();
  // 256 threads: one column each, fixed-order sum over the 8 waves.
  const int col = threadIdx.x;
  float acc = 0.f;
  #pragma unroll
  for (int wv = 0; wv < 8; ++wv) acc += red[wv * 256 + col];
  sPart[((size_t)b * 32 + blockIdx.x) * 256 + col] = acc;
}

// ---------------------------------------------------------------------------
// Kernel 1b: reduce sPart -> sBuf (deterministic fixed order).
// grid (B), block 256.
// ---------------------------------------------------------------------------
__global__ void dl_reduce_s(const float* __restrict__ sPart,
                            float* __restrict__ sBuf) {
  const int b = blockIdx.x, col = threadIdx.x;
  float acc = 0.f;
  #pragma unroll
  for (int i = 0; i < 32; ++i) acc += sPart[((size_t)b * 32 + i) * 256 + col];
  sBuf[b * 256 + col] = acc;
}

// ---------------------------------------------------------------------------
// Kernel 3: WMMA mat-vec dot(row, s_b) + BCE partials.
// grid (32, B), block 256 (8 waves); each wave owns a 16-row tile.
// A = 16x4 fp32 feature chunk, B = s broadcast into all 16 columns, so every
// column of C = dot(row, s). Uses V_WMMA_F32_16X16X4_F32 (exact fp32).
// ---------------------------------------------------------------------------
__global__ void dl_pass2_wmma(const float* __restrict__ features,
                              const float* __restrict__ scores,
                              const float* __restrict__ sBuf,
                              const float* __restrict__ invnorm,
                              float* __restrict__ bcePart) {
  __shared__ float sTile[8 * 16 * 68];  // per-wave 16 rows, stride 68 (bank pad)
  __shared__ float sVec[256];
  __shared__ float yOut[8 * 16];

  const int lane = threadIdx.x & 31;
  const int w    = threadIdx.x >> 5;
  const int b    = blockIdx.y;
  const int r0   = (blockIdx.x * 8 + w) * 16;
  const int m    = lane & 15;       // matrix row within tile
  const int h    = lane >> 4;       // lane half selects K sub-pair
  const float* fB = features + (size_t)b * NN * DD;

  sVec[threadIdx.x] = sBuf[b * 256 + threadIdx.x];

  v8f acc = {};
  #pragma unroll
  for (int c = 0; c < 4; ++c) {         // K chunks of 64
    const int kb0 = c * 64;
    // Stage 16x64 chunk into LDS, coalesced float4.
    #pragma unroll
    for (int it = 0; it < 8; ++it) {
      const int j   = it * 32 + lane;   // 0..255 float4 slots
      const int row = j >> 4;
      const int c4  = j & 15;
      const float4 v = *(const float4*)(fB + (size_t)(r0 + row) * DD + kb0 + c4 * 4);
      *(float4*)&sTile[w * 1088 + row * 68 + c4 * 4] = v;
    }
    __syncthreads();
    #pragma unroll
    for (int g = 0; g < 16; ++g) {      // 16 WMMA K-steps of 4
      const int kl = g * 4;
      // A fragment: lanes 0-15 hold K{0,1}, lanes 16-31 K{2,3} (ISA A layout)
      const v2f a = *(const v2f*)&sTile[w * 1088 + m * 68 + kl + 2 * h];
      // B fragment: s broadcast across all 16 columns, same K split
      const v2f bv = *(const v2f*)&sVec[kb0 + kl + 2 * h];
      acc = __builtin_amdgcn_wmma_f32_16x16x4_f32(
          false, a, false, bv, (short)0, acc, false, false);
    }
    __syncthreads();
  }

  // C layout: VGPR r -> M=r (lanes 0-15) / M=r+8 (lanes 16-31); all N equal.
  if (lane == 0) {
    #pragma unroll
    for (int r = 0; r < 8; ++r) yOut[w * 16 + r] = acc[r];
  } else if (lane == 16) {
    #pragma unroll
    for (int r = 0; r < 8; ++r) yOut[w * 16 + 8 + r] = acc[r];
  }
  __syncthreads();

  float term = 0.f;
  if (lane < 16) {
    const int row  = r0 + lane;
    const float iv = invnorm[b * NN + row];
    const float dotf = yOut[w * 16 + lane] * iv;       // dot(f_row, s_b)
    const float sim  = (dotf - 1.0f) * (1.0f / (float)(NN - 1));
    const float tgt  = 1.0f - fmaxf(sim, 0.0f);
    const float sc   = scores[b * NN + row];
    const float ls   = fmaxf(logf(sc), LOG_CLAMP);
    const float l1   = fmaxf(log1pf(-sc), LOG_CLAMP);
    term = tgt * ls + (1.0f - tgt) * l1;
  }
  // full-wave reduction (lanes >=16 contribute 0)
  term += __shfl_xor(term, 16, 32);
  term += __shfl_xor(term, 8, 32);
  term += __shfl_xor(term, 4, 32);
  term += __shfl_xor(term, 2, 32);
  term += __shfl_xor(term, 1, 32);
  if (lane == 0) {
    const int waveIdx = (blockIdx.y * gridDim.x + blockIdx.x) * 8 + w;
    bcePart[waveIdx] = term;
  }
}

// ---------------------------------------------------------------------------
// Kernel 4: featsum = sum_b ||s_b||^2, bceSum = sum(bcePart); combine.
// grid 1, block 256. Deterministic tree reductions.
// ---------------------------------------------------------------------------
__global__ void dl_final(const float* __restrict__ sBuf,
                         const float* __restrict__ bcePart,
                         float* __restrict__ out) {
  __shared__ float rf[256];
  __shared__ float rb[256];
  const int t = threadIdx.x;
  float fs = 0.f;
  #pragma unroll
  for (int b = 0; b < BB; ++b) {
    const float v = sBuf[b * 256 + t];
    fs += v * v;
  }
  float bs = 0.f;
  #pragma unroll
  for (int i = 0; i < 8; ++i) bs += bcePart[i * 256 + t];
  rf[t] = fs;
  rb[t] = bs;
  __syncthreads();
  for (int off = 128; off > 0; off >>= 1) {
    if (t < off) {
      rf[t] += rf[t + off];
      rb[t] += rb[t + off];
    }
    __syncthreads();
  }
  if (t == 0) {
    const float featsum = rf[0];
    const float bceSum  = rb[0];
    const float feat_loss = 1.0f - featsum * (1.0f / ((float)BB * (float)NN * (float)NN));
    const float bce = -bceSum * (1.0f / ((float)BB * (float)NN));
    out[0] = bce + feat_loss;
  }
}

extern "C" void kernel_launch(void* const* d_in, const int* in_sizes, int n_in,
                              void* d_out, int out_size, void* d_ws, size_t ws_size,
                              hipStream_t stream) {
  (void)in_sizes; (void)n_in; (void)out_size; (void)ws_size;
  const float* features = (const float*)d_in[0];
  const float* scores   = (const float*)d_in[1];
  float* out  = (float*)d_out;
  float* wsF  = (float*)d_ws;
  float* sPart   = wsF + WS_SPART;
  float* sBuf    = wsF + WS_SBUF;
  float* bcePart = wsF + WS_BCEPART;
  float* invnorm = wsF + WS_INVNORM;

  dim3 grid1(32, BB), blk(256);
  dl_pass1<<<grid1, blk, 0, stream>>>(features, sPart, invnorm);
  dl_reduce_s<<<dim3(BB), blk, 0, stream>>>(sPart, sBuf);
  dl_pass2_wmma<<<grid1, blk, 0, stream>>>(features, scores, sBuf, invnorm, bcePart);
  dl_final<<<dim3(1), blk, 0, stream>>>(sBuf, bcePart, out);
}